// AdversarialLoss_16999480557993
// MI455X (gfx1250) — compile-verified
//
#include <hip/hip_runtime.h>

typedef __attribute__((ext_vector_type(2))) float v2f;
typedef __attribute__((ext_vector_type(4))) float v4f;
typedef __attribute__((ext_vector_type(8))) float v8f;

#define CCOLS 32000
#define BLK   320                    // 10 wave32s
#define NWAVE (BLK / 32)             // 10
#define CHUNK 128                    // elements per wave per iteration (32 lanes x float4)
#define NCHUNK (CCOLS / CHUNK)       // 250
#define ITERS (NCHUNK / NWAVE)       // 25 -- exact, no tail

// Numerically stable log(sigmoid(x)) = min(x,0) - log1p(exp(-|x|)).
// __expf/__logf lower to native v_exp_f32 / v_log_f32 sequences.
__device__ __forceinline__ float log_sigmoid(float x) {
    float m = fminf(x, 0.0f);
    float z = __expf(-fabsf(x));
    return m - __logf(1.0f + z);
}

__global__ __launch_bounds__(BLK) void adv_loss_kernel(
    const float* __restrict__ pred,
    const int*   __restrict__ target,
    float*       __restrict__ out)
{
    const int row  = blockIdx.x;
    const int tid  = threadIdx.x;
    const int wave = tid >> 5;
    const int lane = tid & 31;
    const float* rowp = pred + (size_t)row * CCOLS;

    __shared__ float ssum;
    if (tid == 0) ssum = 0.0f;
    __syncthreads();

    // Two independent WMMA accumulators to break the dependent chain.
    v8f acc0 = {};
    v8f acc1 = {};
    v2f ones; ones.x = 1.0f; ones.y = 1.0f;   // B = all-ones 4x16 => D[m,n] = sum_k A[m,k] + C[m,n]

    for (int i = 0; i < ITERS; ++i) {
        const int c = wave + NWAVE * i;                       // chunk id within row
        const v4f* p = (const v4f*)(rowp + c * CHUNK) + lane; // coalesced 512B / wave
        v4f v = __builtin_nontemporal_load(p);                // streaming: don't pollute L2

        v2f a0, a1;
        a0.x = log_sigmoid(v.x);
        a0.y = log_sigmoid(v.y);
        a1.x = log_sigmoid(v.z);
        a1.y = log_sigmoid(v.w);

        // Row-sum on the matrix pipe: each WMMA folds 64 fp32 values into acc.
        acc0 = __builtin_amdgcn_wmma_f32_16x16x4_f32(
                   false, a0, false, ones, (short)0, acc0, false, false);
        acc1 = __builtin_amdgcn_wmma_f32_16x16x4_f32(
                   false, a1, false, ones, (short)0, acc1, false, false);
    }

    // Each of the 16 output columns of D holds a full copy of the partial sum,
    // so sum(all 256 accumulator entries) == 16 * (wave's element sum).
    v8f acc = acc0 + acc1;
    float laneSum = acc[0] + acc[1] + acc[2] + acc[3]
                  + acc[4] + acc[5] + acc[6] + acc[7];

    // wave32 butterfly reduction
    for (int off = 16; off > 0; off >>= 1)
        laneSum += __shfl_xor(laneSum, off, 32);

    if (lane == 0)
        atomicAdd(&ssum, laneSum);      // ds_add_f32, one per wave
    __syncthreads();

    if (tid == 0) {
        float total = ssum * (1.0f / 16.0f);   // undo 16x WMMA column replication
        int t = target[row];
        float lt = log_sigmoid(rowp[t]);       // remove the target column's term
        out[row] = -(total - lt) * (1.0f / (float)CCOLS);
    }
}

extern "C" void kernel_launch(void* const* d_in, const int* in_sizes, int n_in,
                              void* d_out, int out_size, void* d_ws, size_t ws_size,
                              hipStream_t stream) {
    const float* pred   = (const float*)d_in[0];
    const int*   target = (const int*)d_in[1];
    float*       out    = (float*)d_out;
    const int B = out_size;    // 4096 rows, one workgroup per row
    adv_loss_kernel<<<B, BLK, 0, stream>>>(pred, target, out);
}